// StaticGNN_49297634624086
// MI455X (gfx1250) — compile-verified
//
#include <hip/hip_runtime.h>
#include <hip/hip_bf16.h>

// ---------------------------------------------------------------------------
// GCNConv (PyG-style) for gfx1250:  out = relu( Â (x W) + b )
//   N=50000 nodes, E=800000 edges, IN=256, OUT=64, all fp32.
// GEMM: native fp32 WMMA (V_WMMA_F32_16X16X4_F32), A tile staged to LDS via
// gfx1250 async global->LDS (ASYNCcnt), bank-conflict-free padded layout.
// ---------------------------------------------------------------------------

#define N_NODES 50000
#define N_EDGES 800000
#define IN_CH   256
#define OUT_CH  64
#define LDS_STRIDE 260   // 256 + 4 pad floats: stride%64==4 -> conflict-free b64

typedef __attribute__((ext_vector_type(2))) float v2f;
typedef __attribute__((ext_vector_type(8))) float v8f;

// --- 1) deg[n] = 1.0 (self loop weight) -----------------------------------
__global__ void k_init_deg(float* __restrict__ deg, int n) {
    int i = blockIdx.x * blockDim.x + threadIdx.x;
    if (i < n) deg[i] = 1.0f;
}

// --- 2) deg[col[e]] += w[e] ------------------------------------------------
__global__ void k_deg_accum(const int* __restrict__ col,
                            const float* __restrict__ w,
                            float* __restrict__ deg, int e) {
    int i = blockIdx.x * blockDim.x + threadIdx.x;
    if (i < e) atomicAdd(&deg[col[i]], w[i]);
}

// --- 3) dinv[n] = deg>0 ? rsqrt(deg) : 0 ----------------------------------
__global__ void k_dinv(const float* __restrict__ deg,
                       float* __restrict__ dinv, int n) {
    int i = blockIdx.x * blockDim.x + threadIdx.x;
    if (i < n) {
        float d = deg[i];
        dinv[i] = (d > 0.0f) ? __frsqrt_rn(d) : 0.0f;
    }
}

// --- 4) h = x @ W  via fp32 WMMA 16x16x4, A tile async-staged in LDS -------
// Block = 128 threads = 4 waves, one 16-row M tile per block (3125*16==50000
// exactly -> EXEC all-ones, required by WMMA). Each wave: one 16-col N tile.
// Stage: 16x256 f32 A tile (16 KB) -> LDS once per block using
//        global_load_async_to_lds_b128 (ASYNCcnt), then s_wait_asynccnt 0.
// A frag (16x4 f32): lane<16 -> row M=lane, K={k,k+1}; lane>=16 -> K={k+2,k+3}
// B frag (4x16 f32): lane<16 -> col N=lane, K={k,k+1}; lane>=16 -> K={k+2,k+3}
// C/D (16x16 f32, 8 VGPRs): vgpr r -> M = r + 8*(lane>=16), N = lane&15
__global__ __launch_bounds__(128) void k_gemm_wmma(
        const float* __restrict__ x,     // [N, 256] row-major
        const float* __restrict__ W,     // [256, 64] row-major
        float* __restrict__ h) {         // [N, 64] row-major
    __shared__ alignas(16) float tileA[16 * LDS_STRIDE];   // ~16.6 KB

    const int tid   = threadIdx.x;
    const int wave  = tid >> 5;                  // 0..3 -> N tile
    const int lane  = tid & 31;
    const int l15   = lane & 15;
    const int khalf = (lane >> 4) * 2;           // 0 or 2
    const int m0    = blockIdx.x * 16;
    const int n0    = wave * 16;

    // ---- async stage of A tile: 16 rows x 1 KB, 16B chunks ---------------
    // 1024 chunks, 128 threads, 8 passes; rows padded in LDS to LDS_STRIDE.
    {
        const unsigned ldsBase = (unsigned)(uintptr_t)(&tileA[0]);
        const int col4 = (tid & 63) * 4;                   // float col of 16B chunk
        const int rhalf = tid >> 6;                        // 0 or 1
#pragma unroll
        for (int it = 0; it < 8; ++it) {
            const int row = it * 2 + rhalf;                // 0..15
            const float* gsrc = x + (size_t)(m0 + row) * IN_CH + col4;
            unsigned dst = ldsBase + (unsigned)((row * LDS_STRIDE + col4) * 4);
            asm volatile("global_load_async_to_lds_b128 %0, %1, off"
                         :: "v"(dst), "v"(gsrc)
                         : "memory");
        }
        asm volatile("s_wait_asynccnt 0x0" ::: "memory");
        __syncthreads();
    }

    const float* arow = &tileA[l15 * LDS_STRIDE + khalf];  // LDS, 8B aligned
    const float* bcol = W + (size_t)khalf * OUT_CH + n0 + l15;

    v8f c = {};
#pragma unroll 8
    for (int k = 0; k < IN_CH; k += 4) {
        v2f a = *(const v2f*)(arow + k);             // ds_load_b64, no conflicts
        v2f b;
        b.x = bcol[(size_t)k * OUT_CH];              // W[k+khalf  ][n]  (L2-hot)
        b.y = bcol[(size_t)k * OUT_CH + OUT_CH];     // W[k+khalf+1][n]
        // 8 args: (neg_a, A, neg_b, B, c_mod, C, reuse_a, reuse_b)
        c = __builtin_amdgcn_wmma_f32_16x16x4_f32(
                false, a, false, b, (short)0, c, false, false);
    }

    const int mrow = m0 + (lane >> 4) * 8;
    float* hp = h + n0 + l15;
#pragma unroll
    for (int r = 0; r < 8; ++r) {
        hp[(size_t)(mrow + r) * OUT_CH] = c[r];
    }
}

// --- 5) norm[e] = dinv[row]*w*dinv[col] -----------------------------------
__global__ void k_norm(const int* __restrict__ row,
                       const int* __restrict__ col,
                       const float* __restrict__ w,
                       const float* __restrict__ dinv,
                       float* __restrict__ nrm, int e) {
    int i = blockIdx.x * blockDim.x + threadIdx.x;
    if (i < e) nrm[i] = dinv[row[i]] * w[i] * dinv[col[i]];
}

// --- 6) self-loop init: out[n][c] = h[n][c] * dinv[n]^2 --------------------
__global__ void k_selfloop(const float* __restrict__ h,
                           const float* __restrict__ dinv,
                           float* __restrict__ out, int total) {
    int i = blockIdx.x * blockDim.x + threadIdx.x;
    if (i < total) {
        float di = dinv[i >> 6];       // node id (64 ch per node)
        out[i] = h[i] * di * di;
    }
}

// --- 7) scatter: out[col[e]][c] += norm[e] * h[row[e]][c] ------------------
// 64 threads per edge -> fully coalesced 256B reads/atomics per edge.
__global__ void k_scatter(const int* __restrict__ row,
                          const int* __restrict__ col,
                          const float* __restrict__ nrm,
                          const float* __restrict__ h,
                          float* __restrict__ out, int e) {
    int t = blockIdx.x * blockDim.x + threadIdx.x;
    int eidx = t >> 6;
    int c    = t & 63;
    if (eidx < e) {
        int   r  = row[eidx];
        int   d  = col[eidx];
        float nm = nrm[eidx];
        atomicAdd(&out[(size_t)d * OUT_CH + c],
                  h[(size_t)r * OUT_CH + c] * nm);   // global_atomic_add_f32
    }
}

// --- 8) out = relu(out + b) ------------------------------------------------
__global__ void k_finalize(float* __restrict__ out,
                           const float* __restrict__ b, int total) {
    int i = blockIdx.x * blockDim.x + threadIdx.x;
    if (i < total) {
        float v = out[i] + b[i & 63];
        out[i] = v > 0.0f ? v : 0.0f;
    }
}

// ---------------------------------------------------------------------------
extern "C" void kernel_launch(void* const* d_in, const int* in_sizes, int n_in,
                              void* d_out, int out_size, void* d_ws, size_t ws_size,
                              hipStream_t stream) {
    const float* x   = (const float*)d_in[0];            // [N, 256]
    const int*   ei  = (const int*)d_in[1];              // [2, E] flattened
    const float* ew  = (const float*)d_in[2];            // [E]
    const float* W   = (const float*)d_in[3];            // [256, 64]
    const float* b   = (const float*)d_in[4];            // [64]
    float*       out = (float*)d_out;                    // [N, 64]

    const int* row = ei;                                 // source
    const int* col = ei + N_EDGES;                       // target

    // Workspace layout (bytes):
    //   deg  : N_NODES f32            @ 0
    //   dinv : N_NODES f32            @ 0x40000   (256 KiB)
    //   h    : N_NODES*64 f32         @ 0x80000   (512 KiB)
    //   norm : N_EDGES f32            @ 0xD00000  (13 MiB)
    char* ws = (char*)d_ws;
    float* deg  = (float*)(ws);
    float* dinv = (float*)(ws + 0x40000);
    float* h    = (float*)(ws + 0x80000);
    float* nrm  = (float*)(ws + 0xD00000);

    const int TPB = 256;
    const int total = N_NODES * OUT_CH;

    k_init_deg <<<(N_NODES + TPB - 1) / TPB, TPB, 0, stream>>>(deg, N_NODES);
    k_deg_accum<<<(N_EDGES + TPB - 1) / TPB, TPB, 0, stream>>>(col, ew, deg, N_EDGES);
    k_dinv     <<<(N_NODES + TPB - 1) / TPB, TPB, 0, stream>>>(deg, dinv, N_NODES);

    k_gemm_wmma<<<N_NODES / 16, 128, 0, stream>>>(x, W, h);   // 3125 blocks

    k_norm     <<<(N_EDGES + TPB - 1) / TPB, TPB, 0, stream>>>(row, col, ew, dinv, nrm, N_EDGES);
    k_selfloop <<<(total + TPB - 1) / TPB, TPB, 0, stream>>>(h, dinv, out, total);

    {   // 64 threads per edge
        long long threads = (long long)N_EDGES * OUT_CH;
        int blocks = (int)((threads + TPB - 1) / TPB);
        k_scatter<<<blocks, TPB, 0, stream>>>(row, col, nrm, h, out, N_EDGES);
    }

    k_finalize <<<(total + TPB - 1) / TPB, TPB, 0, stream>>>(out, b, total);
}